// Model_Ours_58909771432549
// MI455X (gfx1250) — compile-verified
//
#include <hip/hip_runtime.h>
#include <math.h>
#include <stdint.h>

// Problem constants (match reference)
#define B_    128
#define C_    2048
#define F_    512
#define CLS_  200
#define HW_   14
#define EPS_  1e-5f

typedef float v2f __attribute__((ext_vector_type(2)));
typedef float v8f __attribute__((ext_vector_type(8)));

// =====================================================================
// Stage 1: fused per-(b,c) reductions over the 14x14 spatial tile:
//   - global max pool (mp)
//   - adaptive-avg-pool at scales {4,5,7,11} followed by spatial max (ms)
// One thread owns one (b,c) tile; 64 tiles staged per block through LDS.
// Staging uses CDNA5 GLOBAL_LOAD_ASYNC_TO_LDS_B128: 16B/lane direct
// memory->LDS DMA (no VGPR landing, no ds_store), tracked on ASYNCcnt.
// Tile = 196 floats = 49 float4 -> B128 granularity divides tiles
// exactly; LDS stride 197 floats (gcd(197%64,64)=1) keeps the
// per-thread strided LDS reads bank-conflict free, and padding only
// falls between tiles so each 16B async copy is contiguous in LDS.
// =====================================================================
#define TILE_ELEMS 196
#define TILE_VEC4  49
#define LDS_STRIDE 197
#define RED_TPB    64

__global__ void reduce_ms_kernel(const float* __restrict__ x,
                                 float* __restrict__ mp,
                                 float* __restrict__ ms) {
  __shared__ float lds[RED_TPB * LDS_STRIDE];
  const int tid = threadIdx.x;
  const long long base = (long long)blockIdx.x * (RED_TPB * TILE_ELEMS);

  // Async global -> LDS staging (coalesced: consecutive lanes fetch
  // consecutive 16B chunks of the block's contiguous 12544-float span).
  for (int i = tid; i < RED_TPB * TILE_VEC4; i += RED_TPB) {
    const int t  = i / TILE_VEC4;            // destination tile
    const int o4 = i - t * TILE_VEC4;        // float4 offset inside tile
    const float* gsrc = x + base + (long long)i * 4;
    // Low 32 bits of the generic shared pointer == LDS byte offset.
    unsigned ldsoff = (unsigned)(uintptr_t)(&lds[t * LDS_STRIDE + o4 * 4]);
    asm volatile("global_load_async_to_lds_b128 %0, %1, off"
                 :
                 : "v"(ldsoff), "v"(gsrc)
                 : "memory");
  }
#if __has_builtin(__builtin_amdgcn_s_wait_asynccnt)
  __builtin_amdgcn_s_wait_asynccnt(0);
#else
  asm volatile("s_wait_asynccnt 0" ::: "memory");
#endif
  __syncthreads();

  const int bc = blockIdx.x * RED_TPB + tid;
  const float* tp = &lds[tid * LDS_STRIDE];

  // global max pool
  float m = -3.402823466e38f;
  #pragma unroll 4
  for (int i = 0; i < TILE_ELEMS; ++i) m = fmaxf(m, tp[i]);
  mp[bc] = m;

  // multi-scale adaptive avg pool -> max  (PyTorch bin edges, bins overlap)
  const int scales[4] = {4, 5, 7, 11};
  #pragma unroll
  for (int s = 0; s < 4; ++s) {
    const int k = scales[s];
    float smax = -3.402823466e38f;
    for (int i = 0; i < k; ++i) {
      const int rs = (i * HW_) / k;
      const int re = ((i + 1) * HW_ + k - 1) / k;
      for (int j = 0; j < k; ++j) {
        const int cs = (j * HW_) / k;
        const int ce = ((j + 1) * HW_ + k - 1) / k;
        float sum = 0.f;
        for (int r = rs; r < re; ++r)
          for (int c = cs; c < ce; ++c)
            sum += tp[r * HW_ + c];
        smax = fmaxf(smax, sum / (float)((re - rs) * (ce - cs)));
      }
    }
    ms[(long long)bc * 4 + s] = smax;
  }
}

// =====================================================================
// Per-column batch statistics (train-mode BN, biased variance)
// =====================================================================
__global__ void colstats_kernel(const float* __restrict__ Y,
                                float* __restrict__ mean,
                                float* __restrict__ var,
                                int Bn, int N) {
  const int j = blockIdx.x * blockDim.x + threadIdx.x;
  if (j >= N) return;
  float s = 0.f, s2 = 0.f;
  for (int b = 0; b < Bn; ++b) {
    float v = Y[(long long)b * N + j];
    s += v; s2 += v * v;
  }
  float mu = s / (float)Bn;
  mean[j] = mu;
  var[j] = s2 / (float)Bn - mu * mu;
}

// =====================================================================
// BN-normalize + optional activation.  mode: 0=none, 1=relu, 2=elu
// =====================================================================
__global__ void normact_kernel(const float* __restrict__ Y,
                               const float* __restrict__ mean,
                               const float* __restrict__ var,
                               const float* __restrict__ g,
                               const float* __restrict__ bb,
                               float* __restrict__ outp,
                               int Bn, int N, int mode) {
  const long long idx = (long long)blockIdx.x * blockDim.x + threadIdx.x;
  if (idx >= (long long)Bn * N) return;
  const int j = (int)(idx % N);
  float v = (Y[idx] - mean[j]) * rsqrtf(var[j] + EPS_) * g[j] + bb[j];
  if (mode == 1)      v = fmaxf(v, 0.f);
  else if (mode == 2) v = (v > 0.f) ? v : (expf(v) - 1.f);
  outp[idx] = v;
}

// =====================================================================
// Gate: per-row   elu(BN(mp)) @ g_w.T + g_b -> softmax -> straight-through
// hard one-hot:   gw = (hard - soft) + soft
// One thread per batch row (B=128 -> one block).
// =====================================================================
__global__ void gate_kernel(const float* __restrict__ mp,
                            const float* __restrict__ gm,
                            const float* __restrict__ gv,
                            const float* __restrict__ gg,
                            const float* __restrict__ gb,
                            const float* __restrict__ gW,   // [4, C]
                            const float* __restrict__ gB,   // [4]
                            float* __restrict__ gwOut) {    // [B, 4]
  const int b = threadIdx.x;
  float a0 = 0.f, a1 = 0.f, a2 = 0.f, a3 = 0.f;
  for (int c = 0; c < C_; ++c) {
    float v = (mp[(long long)b * C_ + c] - gm[c]) * rsqrtf(gv[c] + EPS_) * gg[c] + gb[c];
    v = (v > 0.f) ? v : (expf(v) - 1.f);
    a0 += v * gW[0 * C_ + c];
    a1 += v * gW[1 * C_ + c];
    a2 += v * gW[2 * C_ + c];
    a3 += v * gW[3 * C_ + c];
  }
  float l[4] = {a0 + gB[0], a1 + gB[1], a2 + gB[2], a3 + gB[3]};
  float mx = l[0]; int am = 0;
  for (int s = 1; s < 4; ++s) if (l[s] > mx) { mx = l[s]; am = s; }
  float e[4], se = 0.f;
  for (int s = 0; s < 4; ++s) { e[s] = expf(l[s] - mx); se += e[s]; }
  for (int s = 0; s < 4; ++s) {
    float soft = e[s] / se;
    float hard = (s == am) ? 1.f : 0.f;
    gwOut[b * 4 + s] = (hard - soft) + soft;
  }
}

// pool[b,c] = sum_s ms[b,c,s] * gw[b,s]
__global__ void poolcombine_kernel(const float* __restrict__ ms,
                                   const float* __restrict__ gw,
                                   float* __restrict__ pool) {
  const int idx = blockIdx.x * blockDim.x + threadIdx.x;
  if (idx >= B_ * C_) return;
  const int b = idx / C_;
  const float* mrow = ms + (long long)idx * 4;
  const float* grow = gw + b * 4;
  pool[idx] = mrow[0] * grow[0] + mrow[1] * grow[1] +
              mrow[2] * grow[2] + mrow[3] * grow[3];
}

// xf = concat(xl, xh) along channels
__global__ void concat_kernel(const float* __restrict__ xl,
                              const float* __restrict__ xh,
                              float* __restrict__ xf) {
  const int idx = blockIdx.x * blockDim.x + threadIdx.x;
  if (idx >= B_ * 2 * C_) return;
  const int b = idx / (2 * C_);
  const int j = idx - b * (2 * C_);
  xf[idx] = (j < C_) ? xl[(long long)b * C_ + j]
                     : xh[(long long)b * C_ + (j - C_)];
}

// =====================================================================
// Generic fp32 WMMA GEMM:  D[M,N] = A[M,K] @ W[N,K]^T + bias[N]
// One wave32 per 16x16 output tile; K-loop of V_WMMA_F32_16X16X4_F32.
// A-frag (16x4) / B-frag (4x16) lane layout per ISA: lanes 0-15 carry
// K={0,1}, lanes 16-31 carry K={2,3}; C/D row = v + 8*half.
// M is always a multiple of 16 here; N tail (200) guarded per lane.
// =====================================================================
__global__ void wmma_gemm_kernel(const float* __restrict__ A,
                                 const float* __restrict__ W,
                                 const float* __restrict__ bias,
                                 float* __restrict__ D,
                                 int M, int N, int K) {
  const int lane = threadIdx.x & 31;
  const int l = lane & 15;
  const int h = lane >> 4;
  const int wave = blockIdx.x * (blockDim.x >> 5) + (threadIdx.x >> 5);
  const int tilesN = (N + 15) >> 4;
  const int tilesM = (M + 15) >> 4;
  if (wave >= tilesM * tilesN) return;          // wave-uniform exit
  const int tm = (wave / tilesN) << 4;
  const int tn = (wave % tilesN) << 4;

  const int row = tm + l;                       // always < M (M % 16 == 0)
  const int col = tn + l;
  const bool colOK = (col < N);

  const float* ap = A + (long long)row * K + 2 * h;
  const float* wp = W + (long long)(colOK ? col : 0) * K + 2 * h;

  v8f acc = {};
  for (int k = 0; k < K; k += 4) {
    v2f a, b;
    a.x = ap[k];
    a.y = ap[k + 1];
    float b0 = wp[k];
    float b1 = wp[k + 1];
    b.x = colOK ? b0 : 0.f;
    b.y = colOK ? b1 : 0.f;
    __builtin_prefetch(&wp[k + 64], 0, 1);      // global_prefetch_b8 on weights
    acc = __builtin_amdgcn_wmma_f32_16x16x4_f32(false, a, false, b,
                                                (short)0, acc, false, false);
  }

  if (colOK) {
    const float bv = bias[col];
    #pragma unroll
    for (int v = 0; v < 8; ++v) {
      D[(long long)(tm + v + 8 * h) * N + col] = acc[v] + bv;
    }
  }
}

// =====================================================================
// Host-side orchestration
// =====================================================================
static inline void launch_gemm(const float* A, const float* W, const float* bias,
                               float* D, int M, int N, int K, hipStream_t s) {
  const int waves = ((M + 15) / 16) * ((N + 15) / 16);
  const int blocks = (waves + 3) / 4;           // 4 waves (128 thr) per block
  wmma_gemm_kernel<<<blocks, 128, 0, s>>>(A, W, bias, D, M, N, K);
}

static inline void launch_colstats(const float* Y, float* m, float* v,
                                   int Bn, int N, hipStream_t s) {
  colstats_kernel<<<(N + 255) / 256, 256, 0, s>>>(Y, m, v, Bn, N);
}

static inline void launch_normact(const float* Y, const float* m, const float* v,
                                  const float* g, const float* bb, float* o,
                                  int Bn, int N, int mode, hipStream_t s) {
  long long tot = (long long)Bn * N;
  normact_kernel<<<(int)((tot + 255) / 256), 256, 0, s>>>(Y, m, v, g, bb, o, Bn, N, mode);
}

extern "C" void kernel_launch(void* const* d_in, const int* in_sizes, int n_in,
                              void* d_out, int out_size, void* d_ws, size_t ws_size,
                              hipStream_t stream) {
  (void)in_sizes; (void)n_in; (void)out_size; (void)ws_size;

  // ---- inputs (setup_inputs dict order) ----
  const float* x      = (const float*)d_in[0];
  const float* g_bn_g = (const float*)d_in[1];
  const float* g_bn_b = (const float*)d_in[2];
  const float* g_w    = (const float*)d_in[3];
  const float* g_b    = (const float*)d_in[4];
  const float* e_w    = (const float*)d_in[5];
  const float* e_b    = (const float*)d_in[6];
  const float* e_g    = (const float*)d_in[7];
  const float* e_bb   = (const float*)d_in[8];
  const float* const* hl = (const float* const*)&d_in[9];   // g1,b1,w1,bi1,g2,b2,w2,bi2
  const float* const* hh = (const float* const*)&d_in[17];
  const float* const* hf = (const float* const*)&d_in[25];

  float* out = (float*)d_out;

  // ---- workspace layout (floats) ----
  float* ws = (float*)d_ws;
  size_t o = 0;
  float* mp    = ws + o; o += (size_t)B_ * C_;          // [B,C]
  float* ms    = ws + o; o += (size_t)B_ * C_ * 4;      // [B,C,4]
  float* pool  = ws + o; o += (size_t)B_ * C_;          // [B,C]
  float* y1    = ws + o; o += (size_t)B_ * C_;          // [B,C]
  float* xl    = ws + o; o += (size_t)B_ * C_;          // [B,C]
  float* xf    = ws + o; o += (size_t)B_ * 2 * C_;      // [B,2C]
  float* nbuf  = ws + o; o += (size_t)B_ * 2 * C_;      // normalized head input
  float* hbuf  = ws + o; o += (size_t)B_ * 2 * F_;      // head hidden
  float* h2buf = ws + o; o += (size_t)B_ * 2 * F_;      // head hidden (post ELU-BN)
  float* gw    = ws + o; o += (size_t)B_ * 4;           // gate weights
  float* statM = ws + o; o += (size_t)2 * C_;           // mean scratch (<=4096)
  float* statV = ws + o; o += (size_t)2 * C_;           // var scratch  (<=4096)

  // ---- stage 1: fused spatial reductions over x (single pass, HBM-bound) ----
  reduce_ms_kernel<<<(B_ * C_) / RED_TPB, RED_TPB, 0, stream>>>(x, mp, ms);

  // ---- gate ----
  launch_colstats(mp, statM, statV, B_, C_, stream);
  gate_kernel<<<1, B_, 0, stream>>>(mp, statM, statV, g_bn_g, g_bn_b, g_w, g_b, gw);
  poolcombine_kernel<<<(B_ * C_ + 255) / 256, 256, 0, stream>>>(ms, gw, pool);

  // ---- expert_low: Linear(C->C) + BN + ReLU ----
  launch_gemm(pool, e_w, e_b, y1, B_, C_, C_, stream);
  launch_colstats(y1, statM, statV, B_, C_, stream);
  launch_normact(y1, statM, statV, e_g, e_bb, xl, B_, C_, /*relu*/1, stream);

  // ---- xf = concat(xl, mp) ----
  concat_kernel<<<(B_ * 2 * C_ + 255) / 256, 256, 0, stream>>>(xl, mp, xf);

  // ---- heads: BN -> Linear -> BN -> ELU -> Linear ----
  auto run_head = [&](const float* X, int din, int dh,
                      const float* const* p, float* outp) {
    launch_colstats(X, statM, statV, B_, din, stream);
    launch_normact(X, statM, statV, p[0], p[1], nbuf, B_, din, /*none*/0, stream);
    launch_gemm(nbuf, p[2], p[3], hbuf, B_, dh, din, stream);
    launch_colstats(hbuf, statM, statV, B_, dh, stream);
    launch_normact(hbuf, statM, statV, p[4], p[5], h2buf, B_, dh, /*elu*/2, stream);
    launch_gemm(h2buf, p[6], p[7], outp, B_, CLS_, dh, stream);
  };

  run_head(xl, C_,     F_,     hl, out);                       // out_low
  run_head(mp, C_,     F_,     hh, out + (size_t)B_ * CLS_);   // out_high
  run_head(xf, 2 * C_, 2 * F_, hf, out + (size_t)2 * B_ * CLS_); // out
}